// RNNModel_46986942218986
// MI455X (gfx1250) — compile-verified
//
#include <hip/hip_runtime.h>

typedef float v8f __attribute__((ext_vector_type(8)));
typedef float v2f __attribute__((ext_vector_type(2)));

#define DIM_E   64
#define DIM_H   8
#define NLAYER  25
#define NBATCH  1000
#define NT      512
#define TILE_M  16
#define HPAD    12                      // LDS row stride (floats): bank-conflict pad, keeps 16B align
#define NSTEPS  (NT + NLAYER - 1)

// tanh(x) = 1 - 2/(exp(2x)+1);  exp(2x) = exp2(x * 2*log2(e))  -> 1 trans + 1 rcp
__device__ __forceinline__ float fast_tanh(float x) {
  float e = __builtin_amdgcn_exp2f(x * 2.88539008177792681f);
  return 1.0f - 2.0f * __builtin_amdgcn_rcpf(e + 1.0f);
}

// D(16x16 f32) = A(16x4 f32) * B(4x16 f32) + C
__device__ __forceinline__ v8f wmma4(v2f a, v2f b, v8f c) {
  return __builtin_amdgcn_wmma_f32_16x16x4_f32(false, a, false, b, (short)0, c,
                                               false, false);
}

__global__ __launch_bounds__(NLAYER * 32, 1)
void rnn_pipeline_kernel(const int* __restrict__ x,        // [B,T]
                         const float* __restrict__ h0,     // [L,B,H]
                         const float* __restrict__ emb,    // [VOCAB,E]
                         const float* __restrict__ Wxh0,   // [E,H]
                         const float* __restrict__ Wxhr,   // [L-1,H,H]
                         const float* __restrict__ Whh,    // [L,H,H]
                         const float* __restrict__ bh,     // [L,H]
                         const float* __restrict__ Why,    // [H,1]
                         const float* __restrict__ by,     // [1]
                         float* __restrict__ out) {        // [B] logits ++ [L,B,H] h_last
  __shared__ float hbuf[NLAYER][2][TILE_M][HPAD];          // double-buffered layer outputs

  const int tid  = threadIdx.x;
  const int wid  = tid >> 5;            // wave id == layer id (wave32)
  const int lane = tid & 31;
  const int n    = lane & 15;           // N column of B/C fragment (valid when < DIM_H)
  const int kb   = (lane >> 4) << 1;    // K sub-offset within a 4-wide chunk: 0 or 2
  const int tile_base = blockIdx.x * TILE_M;

  // ---- swizzle this layer's weights into WMMA B-fragment registers (once) ----
  v2f  wA[16];                          // layer0: 16 chunks of W_xh0; layers>0: wA[0..3] = [W_xh ; W_hh]
  v2f  wR[2];                           // layer0 only: 2 chunks of W_hh[0]
  float bias = (n < DIM_H) ? bh[wid * DIM_H + n] : 0.0f;

  if (wid == 0) {
#pragma unroll
    for (int c = 0; c < 16; ++c)
#pragma unroll
      for (int j = 0; j < 2; ++j) {
        int k = c * 4 + kb + j;
        wA[c][j] = (n < DIM_H) ? Wxh0[k * DIM_H + n] : 0.0f;
      }
#pragma unroll
    for (int c = 0; c < 2; ++c)
#pragma unroll
      for (int j = 0; j < 2; ++j) {
        int k = c * 4 + kb + j;
        wR[c][j] = (n < DIM_H) ? Whh[k * DIM_H + n] : 0.0f;
      }
  } else {
#pragma unroll
    for (int c = 0; c < 4; ++c)
#pragma unroll
      for (int j = 0; j < 2; ++j) {
        int k = c * 4 + kb + j;
        float v = 0.0f;
        if (n < DIM_H)
          v = (k < DIM_H) ? Wxhr[(wid - 1) * DIM_H * DIM_H + k * DIM_H + n]
                          : Whh[wid * DIM_H * DIM_H + (k - DIM_H) * DIM_H + n];
        wA[c][j] = v;
      }
  }

  // ---- init both parity buffers of this layer's tile from h_0 ----
  if (lane < TILE_M) {
    int gb = tile_base + lane; if (gb > NBATCH - 1) gb = NBATCH - 1;
    const float* hp = h0 + (wid * NBATCH + gb) * DIM_H;
#pragma unroll
    for (int j = 0; j < DIM_H; ++j) {
      float v = hp[j];
      hbuf[wid][0][lane][j] = v;
      hbuf[wid][1][lane][j] = v;
    }
  }
  __syncthreads();

  const int m = n;                      // A-fragment row (batch row in tile)
  int gb = tile_base + m; if (gb > NBATCH - 1) gb = NBATCH - 1;

  // Software-pipelined token index for wave 0: load x[t] one full step ahead so the
  // id->embedding-row dependency chain is hidden behind the previous step's
  // WMMAs + tanh + LDS stores + barrier (no serialized double L2 round-trip).
  int xcur = 0;
  if (wid == 0) xcur = x[gb * NT];      // wave-uniform branch; per-lane row index

  // ---- systolic pipeline over s = t + layer ----
  for (int s = 0; s < NSTEPS; ++s) {
    if (s >= wid && s < wid + NT) {     // wave-uniform: EXEC stays all-ones for WMMA
      const int p_in  = (s + 1) & 1;    // == (s-1)&1
      const int p_out = s & 1;
      v8f acc;
#pragma unroll
      for (int r = 0; r < 8; ++r) acc[r] = bias;

      if (wid == 0) {
        // layer 0: A = embedding[x[:,t]]  (K=64, fragments straight from global/L2)
        const int t = s;
        const float* erow = emb + (long)xcur * DIM_E;
        // issue next step's index load now; its wait lands next iteration
        if (t + 1 < NT) xcur = x[gb * NT + t + 1];
#pragma unroll
        for (int c = 0; c < 16; ++c) {
          v2f a = *(const v2f*)(erow + c * 4 + kb);
          acc = wmma4(a, wA[c], acc);
        }
        // + h_prev @ W_hh[0]  (K=8)
        const float* hp = &hbuf[0][p_in][m][0];
        v2f a0 = *(const v2f*)(hp + kb);
        v2f a1 = *(const v2f*)(hp + 4 + kb);
        acc = wmma4(a0, wR[0], acc);
        acc = wmma4(a1, wR[1], acc);
      } else {
        // fused [inp | h_prev] @ [W_xh ; W_hh]  (K=16 -> 4 WMMAs)
        const float* ip = &hbuf[wid - 1][p_in][m][0];
        const float* hp = &hbuf[wid][p_in][m][0];
        v2f a0 = *(const v2f*)(ip + kb);
        v2f a1 = *(const v2f*)(ip + 4 + kb);
        v2f a2 = *(const v2f*)(hp + kb);
        v2f a3 = *(const v2f*)(hp + 4 + kb);
        acc = wmma4(a0, wA[0], acc);
        acc = wmma4(a1, wA[1], acc);
        acc = wmma4(a2, wA[2], acc);
        acc = wmma4(a3, wA[3], acc);
      }

#pragma unroll
      for (int r = 0; r < 8; ++r) acc[r] = fast_tanh(acc[r]);

      // C layout: VGPR r = row (r | rowbase), col n  -> scatter valid cols to LDS
      if (n < DIM_H) {
        const int rowbase = (lane >> 4) << 3;   // 0 or 8
#pragma unroll
        for (int r = 0; r < 8; ++r)
          hbuf[wid][p_out][rowbase + r][n] = acc[r];
      }
    }
    __syncthreads();
  }

  // ---- write h_last (all waves) and logits (wave L-1) ----
  const int pf = (wid + NT - 1) & 1;
  if (lane < TILE_M) {
    int g = tile_base + lane;
    if (g < NBATCH) {
      const float* src = &hbuf[wid][pf][lane][0];
      float4 v0 = *(const float4*)src;
      float4 v1 = *(const float4*)(src + 4);
      float* dst = out + NBATCH + ((long)wid * NBATCH + g) * DIM_H;
      *(float4*)dst       = v0;
      *(float4*)(dst + 4) = v1;
      if (wid == NLAYER - 1) {
        float a = by[0];
#pragma unroll
        for (int j = 0; j < DIM_H; ++j) a += src[j] * Why[j];
        out[g] = a;                     // logits
      }
    }
  }
}

extern "C" void kernel_launch(void* const* d_in, const int* in_sizes, int n_in,
                              void* d_out, int out_size, void* d_ws, size_t ws_size,
                              hipStream_t stream) {
  (void)in_sizes; (void)n_in; (void)d_ws; (void)ws_size; (void)out_size;
  const int*   x    = (const int*)  d_in[0];
  const float* h0   = (const float*)d_in[1];
  const float* emb  = (const float*)d_in[2];
  const float* Wxh0 = (const float*)d_in[3];
  const float* Wxhr = (const float*)d_in[4];
  const float* Whh  = (const float*)d_in[5];
  const float* bh   = (const float*)d_in[6];
  const float* Why  = (const float*)d_in[7];
  const float* by   = (const float*)d_in[8];
  float* out = (float*)d_out;

  dim3 grid((NBATCH + TILE_M - 1) / TILE_M);   // 63 batch tiles
  dim3 block(NLAYER * 32);                     // one wave32 per layer (systolic pipeline)
  rnn_pipeline_kernel<<<grid, block, 0, stream>>>(x, h0, emb, Wxh0, Wxhr, Whh,
                                                  bh, Why, by, out);
}